// LightGCN_17111149707404
// MI455X (gfx1250) — compile-verified
//
#include <hip/hip_runtime.h>

#define N_USERS 100000
#define N_ITEMS 50000
#define N_NODES (N_USERS + N_ITEMS)   // 150000
#define EMB_DIM 128
#define NNZ     4800000
#define N_LAYERS 3

typedef float v2f __attribute__((ext_vector_type(2)));
typedef float v8f __attribute__((ext_vector_type(8)));

// ---------------------------------------------------------------------------
// init: emb_cur = concat(user_table, item_table); acc (= d_out) = emb_cur
// float4 vectorized: 32 float4 per node row.
// ---------------------------------------------------------------------------
__global__ void lgcn_init(const float4* __restrict__ user,
                          const float4* __restrict__ item,
                          float4* __restrict__ cur,
                          float4* __restrict__ acc) {
    int i = blockIdx.x * blockDim.x + threadIdx.x;       // float4 index
    if (i >= N_NODES * (EMB_DIM / 4)) return;
    int node = i >> 5;                                   // /32 float4 per row
    int c    = i & 31;
    float4 v = (node < N_USERS) ? user[node * 32 + c]
                                : item[(node - N_USERS) * 32 + c];
    cur[i] = v;
    acc[i] = v;
}

__global__ void lgcn_zero(float4* __restrict__ p) {
    int i = blockIdx.x * blockDim.x + threadIdx.x;
    if (i >= N_NODES * (EMB_DIM / 4)) return;
    p[i] = make_float4(0.f, 0.f, 0.f, 0.f);
}

// ---------------------------------------------------------------------------
// SpMM layer: one 32-lane wave per edge. Lane l handles columns 4l..4l+3.
// Edge index is wave-uniform (forced via readfirstlane -> scalar loads stream
// the 57.6MB edge arrays through K$ instead of polluting vector path).
// Gather is a fully-coalesced 512B float4 burst per edge (L2-resident buffer),
// scatter is 4 non-returning global_atomic_add_f32 per lane (resolves in L2).
// ---------------------------------------------------------------------------
__global__ void lgcn_spmm(const int*   __restrict__ rows,
                          const int*   __restrict__ cols,
                          const float* __restrict__ vals,
                          const float* __restrict__ cur,
                          float*       __restrict__ nxt) {
    int t = blockIdx.x * blockDim.x + threadIdx.x;
    int e = __builtin_amdgcn_readfirstlane(t >> 5);      // wave-uniform edge id
    if (e >= NNZ) return;                                // uniform branch
    int lane = threadIdx.x & 31;

    int   r = rows[e];
    int   c = cols[e];
    float v = vals[e];

    const float4* src = (const float4*)(cur + (size_t)c * EMB_DIM) + lane;
    float4 g = *src;                                     // global_load_b128

    float* dst = nxt + (size_t)r * EMB_DIM + lane * 4;
    unsafeAtomicAdd(dst + 0, g.x * v);
    unsafeAtomicAdd(dst + 1, g.y * v);
    unsafeAtomicAdd(dst + 2, g.z * v);
    unsafeAtomicAdd(dst + 3, g.w * v);
}

__global__ void lgcn_accum(float4* __restrict__ acc,
                           const float4* __restrict__ nxt) {
    int i = blockIdx.x * blockDim.x + threadIdx.x;
    if (i >= N_NODES * (EMB_DIM / 4)) return;
    float4 a = acc[i];
    float4 b = nxt[i];
    a.x += b.x; a.y += b.y; a.z += b.z; a.w += b.w;
    acc[i] = a;
}

// ---------------------------------------------------------------------------
// finalize: out = acc * 0.25 via V_WMMA_F32_16X16X4_F32 (exact in fp32).
// One wave handles a 16-row x 4-col chunk:
//   D(16x16) = A(16x4) * B(4x16) + 0, with B = 0.25 * [I4 | 0] so
//   D[:,0:4] = 0.25 * A.
// Lane layouts per ISA 7.12.2 (wave32):
//   A: lanes 0-15 hold M=lane, VGPR0=K0/VGPR1=K1; lanes 16-31 hold K2/K3.
//   B: VGPR0 = row K (0 or 2 by lane half), N = lane&15.
//   D: VGPR j = row M=j (lanes 0-15) / M=j+8 (lanes 16-31), N = lane&15.
// EXEC is all-1s at the WMMA; only the store afterwards is predicated.
// Grid is exact (150000/16 * 32 waves = 300000 = 37500 blocks * 8 waves).
// ---------------------------------------------------------------------------
__global__ void lgcn_finalize_wmma(float* __restrict__ out) {
    const int wave = blockIdx.x * (blockDim.x >> 5) + (threadIdx.x >> 5);
    const int lane = threadIdx.x & 31;

    const int r0 = (wave >> 5) * 16;      // 32 col-chunks per 16-row tile
    const int c0 = (wave & 31) * 4;

    const int m    = lane & 15;
    const int koff = (lane < 16) ? 0 : 2;

    // A operand: two consecutive columns of acc for row (r0+m)
    const float2* ap =
        (const float2*)(out + (size_t)(r0 + m) * EMB_DIM + c0 + koff);
    float2 av = *ap;
    v2f A; A[0] = av.x; A[1] = av.y;

    // B operand: 0.25 * selection matrix (rows koff, koff+1; col = m)
    const float s = 1.0f / (float)(N_LAYERS + 1);        // 0.25
    v2f B;
    B[0] = (m == koff)     ? s : 0.0f;
    B[1] = (m == koff + 1) ? s : 0.0f;

    v8f C = {};   // zero accumulator
    v8f D = __builtin_amdgcn_wmma_f32_16x16x4_f32(
        /*neg_a=*/false, A, /*neg_b=*/false, B,
        /*c_mod=*/(short)0, C, /*reuse_a=*/false, /*reuse_b=*/false);

    // Only columns 0..3 of D are meaningful: lanes with N<4 store them.
    if (m < 4) {
        const int col = c0 + m;
        const int rb  = r0 + ((lane < 16) ? 0 : 8);
#pragma unroll
        for (int j = 0; j < 8; ++j)
            out[(size_t)(rb + j) * EMB_DIM + col] = D[j];
    }
}

// ---------------------------------------------------------------------------
extern "C" void kernel_launch(void* const* d_in, const int* in_sizes, int n_in,
                              void* d_out, int out_size, void* d_ws, size_t ws_size,
                              hipStream_t stream) {
    const float4* user = (const float4*)d_in[0];
    const float4* item = (const float4*)d_in[1];
    const int*    rows = (const int*)d_in[2];
    const int*    cols = (const int*)d_in[3];
    const float*  vals = (const float*)d_in[4];
    float* out = (float*)d_out;

    const size_t emb_elems = (size_t)N_NODES * EMB_DIM;  // 19.2M floats
    float* buf0 = (float*)d_ws;
    float* buf1 = buf0 + emb_elems;

    const int vec_threads = N_NODES * (EMB_DIM / 4);     // 4.8M float4
    const int vec_blocks  = (vec_threads + 255) / 256;   // 18750

    const int spmm_blocks = (int)(((long long)NNZ * 32 + 255) / 256); // 600000

    lgcn_init<<<vec_blocks, 256, 0, stream>>>(user, item,
                                              (float4*)buf0, (float4*)out);

    float* cur = buf0;
    float* nxt = buf1;
    for (int l = 0; l < N_LAYERS; ++l) {
        lgcn_zero<<<vec_blocks, 256, 0, stream>>>((float4*)nxt);
        lgcn_spmm<<<spmm_blocks, 256, 0, stream>>>(rows, cols, vals, cur, nxt);
        lgcn_accum<<<vec_blocks, 256, 0, stream>>>((float4*)out,
                                                   (const float4*)nxt);
        float* tmp = cur; cur = nxt; nxt = tmp;
    }

    // 300000 waves total, 8 waves per 256-thread block -> 37500 blocks exact.
    lgcn_finalize_wmma<<<37500, 256, 0, stream>>>(out);
}